// FlowCPAB_22952305229988
// MI455X (gfx1250) — compile-verified
//
#include <hip/hip_runtime.h>
#include <math.h>

#define TESS   16
#define NBASIS 17          // 2*TESS - rank(L) = 32 - 15
#define NPARAM 32          // 2*TESS
#define EPSF   1e-7f

typedef __attribute__((ext_vector_type(2))) float v2f;
typedef __attribute__((ext_vector_type(4))) float f4;
typedef __attribute__((ext_vector_type(8))) float v8f;

__device__ __forceinline__ double wave_sum_f64(double v) {
  // wave32 butterfly reduction
  #pragma unroll
  for (int o = 16; o > 0; o >>= 1) v += __shfl_xor(v, o, 32);
  return v;
}

// Null-space spanning vector i (i<16: a=e_i, b0=0; i==16: a=0, b=1),
// component p of the 32-vector (a0,b0,a1,b1,...).
__device__ __forceinline__ double nullspace_comp(int i, int p) {
  int k = p >> 1;
  if ((p & 1) == 0) return (i < TESS && k == i) ? 1.0 : 0.0;
  if (i == TESS) return 1.0;
  double v = 0.0;
  if (i + 1 <= k) v += (double)(i + 1) / TESS;     // +x_{i+1} once k >= i+1
  if (i >= 1 && i <= k) v -= (double)i / TESS;     // -x_i once k >= i
  return v;
}

// ---------------------------------------------------------------------------
// Kernel 1 (1 wave): build an orthonormal basis of null(L) (deterministic
// f64 modified Gram-Schmidt; NOTE: numpy-SVD's particular basis is not
// reproducible on device and B is not a kernel input), then compute
// A = theta @ B^T exactly with V_WMMA_F32_16X16X4_F32 (f32 inputs => exact).
// dAB[2c] = a_c, dAB[2c+1] = b_c.
// ---------------------------------------------------------------------------
__global__ __launch_bounds__(32)
void cpab_setup_kernel(const float* __restrict__ theta, float* __restrict__ dAB) {
  __shared__ double Vs[NBASIS][32];   // Vs[i][p] = component p of basis vec i
  const int lane = threadIdx.x;       // lane == component index p

  for (int i = 0; i < NBASIS; ++i) Vs[i][lane] = nullspace_comp(i, lane);
  __syncthreads();

  // Lane-parallel modified Gram-Schmidt (each lane owns component `lane`).
  for (int i = 0; i < NBASIS; ++i) {
    double vi = Vs[i][lane];
    for (int j = 0; j < i; ++j) {
      double vj = Vs[j][lane];
      double d  = wave_sum_f64(vi * vj);
      vi -= d * vj;
    }
    double n2 = wave_sum_f64(vi * vi);
    vi *= 1.0 / sqrt(n2);
    Vs[i][lane] = vi;
  }
  __syncthreads();

  // WMMA GEMV: out[j] = sum_k theta[k] * Vs[k][j], j = 0..31.
  // Theta replicated across all 16 M-rows -> every row of D equals out.
  // A-matrix 16x4 f32 layout: lanes 0-15 hold K=0,1 ; lanes 16-31 hold K=2,3.
  // B-matrix 4x16 layout:     VGPR0 = rows K=0 (lanes 0-15) / K=2 (16-31),
  //                           VGPR1 = rows K=1 / K=3.
  const int khalf = (lane >> 4) << 1;
  const int n     = lane & 15;

  v8f accLo = {};   // columns 0..15  of out
  v8f accHi = {};   // columns 16..31 of out
  #pragma unroll
  for (int k0 = 0; k0 < 20; k0 += 4) {     // K padded 17 -> 20
    const int ka = k0 + khalf, kb = k0 + khalf + 1;
    v2f tA, tBlo, tBhi;
    tA.x   = (ka < NBASIS) ? theta[ka] : 0.0f;
    tA.y   = (kb < NBASIS) ? theta[kb] : 0.0f;
    tBlo.x = (ka < NBASIS) ? (float)Vs[ka][n]      : 0.0f;
    tBlo.y = (kb < NBASIS) ? (float)Vs[kb][n]      : 0.0f;
    tBhi.x = (ka < NBASIS) ? (float)Vs[ka][16 + n] : 0.0f;
    tBhi.y = (kb < NBASIS) ? (float)Vs[kb][16 + n] : 0.0f;
    accLo = __builtin_amdgcn_wmma_f32_16x16x4_f32(false, tA, false, tBlo,
                                                  (short)0, accLo, false, false);
    accHi = __builtin_amdgcn_wmma_f32_16x16x4_f32(false, tA, false, tBhi,
                                                  (short)0, accHi, false, false);
  }
  // D[m][n] identical over m; lane L<16 holds col L, lane L>=16 holds col L-16
  // of each half -> dAB[lane] works for both halves.
  dAB[lane] = (lane < 16) ? accLo[0] : accHi[0];
}

// ---------------------------------------------------------------------------
// Kernel 2: streaming closed-form CPAB integration. HBM-bandwidth bound:
// 128-bit nontemporal loads/stores, (a,b) pairs via one ds_load_b64 per hop,
// one v_log_f32 per hop, single v_exp_f32 at the final partial step.
// ---------------------------------------------------------------------------
__device__ __forceinline__ void integrate_one(float x0, const float2* sAB,
                                              float& z, float& ph) {
  const float h = 1.0f / TESS;
  float xv = fminf(fmaxf(x0, EPSF), 1.0f - EPSF);
  int   c  = (int)floorf(xv * (float)TESS);
  c = min(max(c, 0), TESS - 1);
  float t = 1.0f, phase = 0.0f;

  #pragma unroll 1
  for (int it = 0; it < TESS; ++it) {
    float2 ab = sAB[c];
    const float ac = ab.x, bc = ab.y;
    const float v = fmaf(ac, xv, bc);
    const bool  right    = (v >= 0.0f);
    const bool  edge_out = (right && c == TESS - 1) || (!right && c == 0);
    const float xb = (right ? (float)(c + 1) : (float)c) * h;
    const float vb = fmaf(ac, xb, bc);
    const bool  a_zero = fabsf(ac) < 1e-12f;
    const bool  v_zero = fabsf(v)  < 1e-15f;
    const float ac_safe = a_zero ? 1.0f : ac;
    const float v_safe  = v_zero ? 1.0f : v;
    const float ratio   = __fdividef(vb, v_safe);
    float thit = a_zero ? __fdividef(xb - xv, v_safe)
                        : __fdividef(__logf(fmaxf(ratio, 1e-30f)), ac_safe);
    const bool reachable = (!edge_out) && (!v_zero) && (ratio > 0.0f);
    if (!reachable) thit = __builtin_inff();

    if (thit <= t) {                       // exit this cell
      phase = fmaf(ac, thit, phase);       // == log(ratio) in affine case
      xv = xb;
      c  = min(max(c + (right ? 1 : -1), 0), TESS - 1);
      t -= thit;
    } else {                               // settle in this cell: dt = t
      const float ooa = __fdividef(bc, ac_safe);
      const float e   = __expf(ac * t);
      xv    = a_zero ? fmaf(v, t, xv) : fmaf(xv + ooa, e, -ooa);
      phase = fmaf(ac, t, phase);
      break;
    }
  }
  z = xv; ph = phase;
}

__global__ __launch_bounds__(256)
void cpab_integrate_kernel(const float* __restrict__ x,
                           const float* __restrict__ dAB,
                           float* __restrict__ z_out,
                           float* __restrict__ phase_out,
                           int n) {
  __shared__ float2 sAB[TESS];
  if (threadIdx.x < TESS)
    sAB[threadIdx.x] = make_float2(dAB[2 * threadIdx.x], dAB[2 * threadIdx.x + 1]);
  __syncthreads();

  const int i4   = blockIdx.x * blockDim.x + threadIdx.x;
  const int base = i4 * 4;
  if (base >= n) return;

  if (base + 4 <= n) {
    const f4 xin = __builtin_nontemporal_load((const f4*)x + i4);
    float z0, z1, z2, z3, p0, p1, p2, p3;
    integrate_one(xin.x, sAB, z0, p0);
    integrate_one(xin.y, sAB, z1, p1);
    integrate_one(xin.z, sAB, z2, p2);
    integrate_one(xin.w, sAB, z3, p3);
    f4 zo, po;
    zo.x = z0; zo.y = z1; zo.z = z2; zo.w = z3;
    po.x = p0; po.y = p1; po.z = p2; po.w = p3;
    __builtin_nontemporal_store(zo, (f4*)z_out + i4);
    __builtin_nontemporal_store(po, (f4*)phase_out + i4);
  } else {
    for (int k = base; k < n; ++k) {
      float zv, pv;
      integrate_one(x[k], sAB, zv, pv);
      z_out[k] = zv; phase_out[k] = pv;
    }
  }
}

extern "C" void kernel_launch(void* const* d_in, const int* in_sizes, int n_in,
                              void* d_out, int out_size, void* d_ws, size_t ws_size,
                              hipStream_t stream) {
  const float* x     = (const float*)d_in[0];
  const float* theta = (const float*)d_in[1];
  const int n = in_sizes[0];            // x is [N,1] -> flat N

  float* out       = (float*)d_out;     // tuple (z, phase) concatenated
  float* z_out     = out;
  float* phase_out = out + n;
  float* dAB       = (float*)d_ws;      // 32 floats: (a_c,b_c) interleaved

  cpab_setup_kernel<<<1, 32, 0, stream>>>(theta, dAB);

  const int nthreads = (n + 3) / 4;
  const int blocks   = (nthreads + 255) / 256;
  cpab_integrate_kernel<<<blocks, 256, 0, stream>>>(x, dAB, z_out, phase_out, n);
}